// Q_conv_29618094474041
// MI455X (gfx1250) — compile-verified
//
#include <hip/hip_runtime.h>
#include <hip/hip_bf16.h>

// Quanvolution on MI455X (gfx1250).
//
// Restructured as:  ev = Zreduce( | m @ W' |^2 )
//   m  : (P=31752, 512) real, per-patch product-state magnitudes
//   W' : (512, 512) complex, circuit unitary with (-i)^popcount phase folded in
// The big GEMM runs on V_WMMA_F32_16X16X4_F32 (full fp32 precision; the
// amplitudes are O(1/sqrt(512)) so low-precision WMMA would corrupt results).
//
// d_ws usage: B_re (512*512 f32) + B_im (512*512 f32) = 2 MB.

#define NQ      9
#define DIM     512           // 2^9 amplitudes
#define PATCHES 31752         // 8 * 63 * 63
#define OUTW    63
#define OO      3969          // 63*63
#define MST     516           // padded LDS row stride (floats): 516*4 % 256 = 16 -> conflict-free column reads

typedef __attribute__((ext_vector_type(2))) float v2f;
typedef __attribute__((ext_vector_type(8))) float v8f;

// ---------------------------------------------------------------------------
// Kernel 1: build W' rows.  Block a evolves basis state |a> through the fixed
// circuit (weights-dependent only), folds (-i)^popcount(a), stores row a of
// B_re/B_im in k-pair-interleaved layout:  B[(k>>1)*1024 + n*2 + (k&1)]
// so the GEMM's B fragment (K, K+1 for one column n) is one contiguous 8B load.
// Qubit q maps to bit position (8-q) of the flat amplitude index (reference
// uses axis 1+q of a (2,)*9 tensor => qubit 0 is the MSB).
// ---------------------------------------------------------------------------
__global__ __launch_bounds__(256) void build_unitary(
    const float* __restrict__ w,     // (2, 9, 3) = 54 floats
    float* __restrict__ Bre, float* __restrict__ Bim)
{
    __shared__ float re[DIM];
    __shared__ float im[DIM];
    const int a = blockIdx.x;      // initial basis index = K row of the GEMM
    const int t = threadIdx.x;     // 0..255

    re[t]       = (t == a)       ? 1.0f : 0.0f;  im[t]       = 0.0f;
    re[t + 256] = (t + 256 == a) ? 1.0f : 0.0f;  im[t + 256] = 0.0f;
    __syncthreads();

    for (int rep = 0; rep < 2; ++rep) {
        for (int l = 0; l < 2; ++l) {
            // ---- 9 Rot(phi,theta,omega) = RZ(omega) RY(theta) RZ(phi) gates
            for (int q = 0; q < NQ; ++q) {
                const float phi = w[l * 27 + q * 3 + 0];
                const float th  = w[l * 27 + q * 3 + 1];
                const float om  = w[l * 27 + q * 3 + 2];
                float st, ct;  __sincosf(0.5f * th, &st, &ct);
                float sa, ca;  __sincosf(0.5f * (phi + om), &sa, &ca);
                float sb, cb;  __sincosf(0.5f * (phi - om), &sb, &cb);
                // u00 = e^{-i(phi+om)/2} c ; u01 = -e^{+i(phi-om)/2} s
                // u10 = e^{-i(phi-om)/2} s ; u11 = e^{+i(phi+om)/2} c
                const float u00r =  ct * ca, u00i = -ct * sa;
                const float u01r = -st * cb, u01i = -st * sb;
                const float u10r =  st * cb, u10i = -st * sb;
                const float u11r =  ct * ca, u11i =  ct * sa;

                const int pb  = 8 - q;
                const int low = (1 << pb) - 1;
                const int i0  = ((t & ~low) << 1) | (t & low);
                const int i1  = i0 | (1 << pb);
                const float v0r = re[i0], v0i = im[i0];
                const float v1r = re[i1], v1i = im[i1];
                // each thread owns a disjoint (i0,i1) pair -> no intra-gate race
                re[i0] = u00r * v0r - u00i * v0i + u01r * v1r - u01i * v1i;
                im[i0] = u00r * v0i + u00i * v0r + u01r * v1i + u01i * v1r;
                re[i1] = u10r * v0r - u10i * v0i + u11r * v1r - u11i * v1i;
                im[i1] = u10r * v0i + u10i * v0r + u11r * v1i + u11i * v1r;
                __syncthreads();
            }
            // ---- 9 ring CNOTs: control q, target (q + l + 1) % 9, in order
            const int r = l + 1;
            for (int q = 0; q < NQ; ++q) {
                const int cbit = 1 << (8 - q);
                const int tbit = 1 << (8 - ((q + r) % NQ));
                const int a0 = t, a1 = t + 256;
                const int s0 = (a0 & cbit) ? (a0 ^ tbit) : a0;
                const int s1 = (a1 & cbit) ? (a1 ^ tbit) : a1;
                const float r0 = re[s0], q0 = im[s0];
                const float r1 = re[s1], q1 = im[s1];
                __syncthreads();
                re[a0] = r0; im[a0] = q0;
                re[a1] = r1; im[a1] = q1;
                __syncthreads();
            }
        }
    }

    // fold (-i)^popcount(a) into this row, write k-pair-interleaved
    const int ph = __popc(a) & 3;
    const long rowbase = (long)(a >> 1) * (2 * DIM) + (a & 1);
    for (int e = t; e < DIM; e += 256) {
        const float x = re[e], y = im[e];
        float xr, yi;
        if      (ph == 0) { xr =  x; yi =  y; }
        else if (ph == 1) { xr =  y; yi = -x; }
        else if (ph == 2) { xr = -x; yi = -y; }
        else              { xr = -y; yi =  x; }
        Bre[rowbase + (long)e * 2] = xr;
        Bim[rowbase + (long)e * 2] = yi;
    }
}

// ---------------------------------------------------------------------------
// Kernel 2: per block: 16 patches.  Build m[16][512] in LDS, then 8 waves each
// run a 16x64x512 double GEMM (psi_re, psi_im) on v_wmma_f32_16x16x4_f32,
// square into LDS probs, and 144 threads do the signed Z reductions.
// A fragment (16x4 f32):  lane<16: M=lane,    (K0,K1);  lane>=16: M=lane-16, (K2,K3)
// B fragment (4x16 f32):  lane<16: N=lane,    (K0,K1);  lane>=16: N=lane-16, (K2,K3)
// C/D (16x16 f32): VGPR r: lanes 0-15 -> (M=r, N=lane); lanes 16-31 -> (M=r+8)
// ---------------------------------------------------------------------------
__global__ __launch_bounds__(256) void qconv_gemm(
    const float* __restrict__ in,    // (8,1,128,128)
    const float* __restrict__ Bre, const float* __restrict__ Bim,
    float* __restrict__ out)         // (8,9,63,63)
{
    extern __shared__ float smem[];
    float* m  = smem;                // 16 * MST
    float* pr = m + 16 * MST;        // 16 * MST
    float* cc = pr + 16 * MST;       // 16 * 9 cos(theta/2)
    float* ss = cc + 144;            // 16 * 9 sin(theta/2)

    const int tid = threadIdx.x;
    const int p0  = blockIdx.x * 16;

    // ---- per-patch half-angle cos/sin
    if (tid < 144) {
        const int lp = tid / 9, j = tid % 9;
        const int p  = p0 + lp;
        float c = 1.0f, s = 0.0f;
        if (p < PATCHES) {
            const int b  = p / OO,  rem = p % OO;
            const int oy = rem / OUTW, ox = rem % OUTW;
            const int ky = j / 3, kx = j % 3;
            const float ang = in[(long)b * 16384 + (oy * 2 + ky) * 128 + (ox * 2 + kx)];
            __sincosf(0.5f * ang, &s, &c);
        }
        cc[tid] = c; ss[tid] = s;
    }
    __syncthreads();

    // ---- m[lp][a] = prod_q (bit_{8-q}(a) ? sin_q : cos_q)
    for (int e = tid; e < 16 * DIM; e += 256) {
        const int lp = e >> 9, a = e & (DIM - 1);
        float prod = 1.0f;
#pragma unroll
        for (int q = 0; q < NQ; ++q)
            prod *= ((a >> (8 - q)) & 1) ? ss[lp * 9 + q] : cc[lp * 9 + q];
        m[lp * MST + a] = prod;
    }
    __syncthreads();

    // ---- WMMA GEMM: wave w owns N columns [w*64, w*64+64)
    const int wave  = tid >> 5;
    const int lane  = tid & 31;
    const int row   = lane & 15;            // M row (A) / N col (B,C)
    const int koff  = (lane >> 4) << 1;     // 0 or 2
    const int mrow0 = (lane >> 4) << 3;     // C: rows 0..7 or 8..15

    for (int cidx = 0; cidx < 4; ++cidx) {
        const int n0 = wave * 64 + cidx * 16;
        const int n  = n0 + row;
        v8f accr = {};
        v8f acci = {};
        for (int k0 = 0; k0 < DIM; k0 += 4) {
            const int kk = k0 + koff;       // even
            v2f av;
            av.x = m[row * MST + kk];
            av.y = m[row * MST + kk + 1];
            const long bidx = (long)(kk >> 1) * (2 * DIM) + n * 2;
            v2f br; br.x = Bre[bidx]; br.y = Bre[bidx + 1];
            v2f bi; bi.x = Bim[bidx]; bi.y = Bim[bidx + 1];
            accr = __builtin_amdgcn_wmma_f32_16x16x4_f32(
                false, av, false, br, (short)0, accr, false, false);
            acci = __builtin_amdgcn_wmma_f32_16x16x4_f32(
                false, av, false, bi, (short)0, acci, false, false);
        }
        // probabilities |psi|^2 into LDS (each wave owns disjoint columns)
#pragma unroll
        for (int rr = 0; rr < 8; ++rr) {
            const float xr = accr[rr], xi = acci[rr];
            pr[(mrow0 + rr) * MST + n] = xr * xr + xi * xi;
        }
    }
    __syncthreads();

    // ---- PauliZ expectations: ev[p,q] = sum_b (1 - 2*bit_{8-q}(b)) * prob[p,b]
    if (tid < 144) {
        const int lp = tid / 9, q = tid % 9;
        const int p  = p0 + lp;
        if (p < PATCHES) {
            const int pb = 8 - q;
            float acc = 0.0f;
            for (int b = 0; b < DIM; ++b) {
                const float v = pr[lp * MST + b];
                acc += ((b >> pb) & 1) ? -v : v;
            }
            const int bb = p / OO, rem = p % OO;
            out[(long)bb * (NQ * OO) + q * OO + rem] = acc;
        }
    }
}

extern "C" void kernel_launch(void* const* d_in, const int* in_sizes, int n_in,
                              void* d_out, int out_size, void* d_ws, size_t ws_size,
                              hipStream_t stream) {
    const float* input   = (const float*)d_in[0];   // (8,1,128,128)
    const float* weights = (const float*)d_in[1];   // (2,9,3)
    float* Bre = (float*)d_ws;                      // 512*512 f32
    float* Bim = Bre + DIM * DIM;                   // 512*512 f32   (2 MB total)

    build_unitary<<<DIM, 256, 0, stream>>>(weights, Bre, Bim);

    const int nblk = (PATCHES + 15) / 16;           // 1985
    const size_t shmem = (size_t)(2 * 16 * MST + 288) * sizeof(float);  // 67200 B
    qconv_gemm<<<nblk, 256, shmem, stream>>>(input, Bre, Bim, (float*)d_out);
}